// HashEmbedding_16432544874939
// MI455X (gfx1250) — compile-verified
//
#include <hip/hip_runtime.h>
#include <stdint.h>

typedef float v8f  __attribute__((ext_vector_type(8)));
typedef float v2f  __attribute__((ext_vector_type(2)));
typedef unsigned int u32x4 __attribute__((ext_vector_type(4)));
typedef int   i32x4 __attribute__((ext_vector_type(4)));
typedef int   i32x8 __attribute__((ext_vector_type(8)));

#define HIST   200
#define DIM    128          // embedding dim (one row = 512 B)
#define WPB    4            // waves per block (wave32)
#define CHUNK  8            // rows per TDM gather (32-bit indices -> max 8)
#define NCHUNK (HIST / CHUNK)   // 25 gathers per batch element

// Generic pointers into the LDS aperture carry the wave-relative LDS byte
// offset in their low 32 bits (ISA 10.2).
__device__ __forceinline__ uint32_t lds_off(const void* p) {
  return (uint32_t)(uintptr_t)p;
}

__global__ __launch_bounds__(32 * WPB) void hashemb_tdm_wmma_kernel(
    const uint32_t* __restrict__ x32,   // int64 ids viewed as LE uint32 pairs
    const float*    __restrict__ table,
    float*          __restrict__ out,
    uint32_t nrows, uint32_t batch)
{
  __shared__ float smem[WPB * 2 * CHUNK * DIM];   // 4 waves * 2 bufs * 4KB = 32KB

  const int lane = threadIdx.x & 31;
  const int wave = threadIdx.x >> 5;
  const int b    = blockIdx.x * WPB + wave;
  if ((uint32_t)b >= batch) return;               // wave-uniform exit

  float* buf0 = &smem[wave * (2 * CHUNK * DIM)];
  float* buf1 = buf0 + CHUNK * DIM;
  const uint32_t lds0 = lds_off(buf0);
  const uint32_t lds1 = lds_off(buf1);

  // ---- preload & hash all 200 ids; reg j holds history item h = 32*j+lane ----
  uint32_t idxreg[7];
  const uint32_t base = (uint32_t)b * HIST;
  #pragma unroll
  for (int j = 0; j < 7; ++j) {
    const int h = 32 * j + lane;
    uint32_t id = 0;
    if (h < HIST) id = x32[2u * (base + (uint32_t)h)];  // low word of int64
    idxreg[j] = (id * 2654435761u) % nrows;             // Knuth mix -> bucket
  }

  // ---- constant parts of the Tensor-DMA gather descriptor (D#) ----
  const uint64_t gaddr = (uint64_t)(uintptr_t)table;
  u32x4 g0_0, g0_1;
  {
    const uint32_t w0 = 1u | (1u << 30) | (1u << 31);   // count=1 | 32b idx | gather
    const uint32_t w2 = (uint32_t)gaddr;                // global_addr[31:0]
    const uint32_t w3 = (uint32_t)((gaddr >> 32) & 0x01FFFFFFu) | (2u << 30); // [56:32] | type=2
    g0_0[0] = w0; g0_0[1] = lds0; g0_0[2] = w2; g0_0[3] = w3;
    g0_1[0] = w0; g0_1[1] = lds1; g0_1[2] = w2; g0_1[3] = w3;
  }
  i32x8 g1;
  g1[0] = (int)(2u << 16);                     // data_size = 4 B; no multicast
  g1[1] = (int)((uint32_t)DIM << 16);          // tensor_dim0 = 128 (low 16)
  g1[2] = (int)((nrows & 0xFFFFu) << 16);      // dim0 hi16 = 0 | tensor_dim1 lo16
  g1[3] = (int)(((nrows >> 16) & 0xFFFFu) | ((uint32_t)DIM << 16)); // dim1 hi | tile_dim0=128
  g1[4] = CHUNK;                               // tile_dim1 = #valid gather indices
  g1[5] = DIM;                                 // tensor_dim0_stride = 128 elements
  g1[6] = 0;
  g1[7] = 0;

  // Issue one 8-row gather for chunk c into buffer described by g0.
  auto issue = [&](int c, const u32x4& g0) {
    const int j  = c >> 2;            // compile-time under full unroll
    const int l0 = (c & 3) * 8;
    i32x4 g2, g3;
    g2[0] = __builtin_amdgcn_readlane((int)idxreg[j], l0 + 0);
    g2[1] = __builtin_amdgcn_readlane((int)idxreg[j], l0 + 1);
    g2[2] = __builtin_amdgcn_readlane((int)idxreg[j], l0 + 2);
    g2[3] = __builtin_amdgcn_readlane((int)idxreg[j], l0 + 3);
    g3[0] = __builtin_amdgcn_readlane((int)idxreg[j], l0 + 4);
    g3[1] = __builtin_amdgcn_readlane((int)idxreg[j], l0 + 5);
    g3[2] = __builtin_amdgcn_readlane((int)idxreg[j], l0 + 6);
    g3[3] = __builtin_amdgcn_readlane((int)idxreg[j], l0 + 7);
#if __clang_major__ >= 23
    i32x8 zz = {0, 0, 0, 0, 0, 0, 0, 0};
    __builtin_amdgcn_tensor_load_to_lds(g0, g1, g2, g3, zz, 0);
#else
    __builtin_amdgcn_tensor_load_to_lds(g0, g1, g2, g3, 0);
#endif
  };

  // ---- accumulators: 8 column-chunks of a 16x16 f32 D-matrix (all rows equal) ----
  v8f acc[8];
  #pragma unroll
  for (int cc = 0; cc < 8; ++cc) {
    v8f z = {0.f, 0.f, 0.f, 0.f, 0.f, 0.f, 0.f, 0.f};
    acc[cc] = z;
  }
  const v2f ones = {1.0f, 1.0f};    // A (16x4) = ones -> D[m,n] = C + sum_k B[k,n]

  const int n    = lane & 15;       // B-matrix column handled by this lane
  const int half = lane >> 4;       // lanes 16-31 hold the upper K pair

  // WMMA-accumulate the 8 rows of chunk c sitting in LDS buffer buf[c&1].
  auto consume = [&](int c) {
    const float* buf = (c & 1) ? buf1 : buf0;   // 8 rows x 128 cols, row-major
    #pragma unroll
    for (int g = 0; g < 2; ++g) {               // K-groups: rows 4g..4g+3
      #pragma unroll
      for (int cc = 0; cc < 8; ++cc) {          // 16-column chunks
        // B layout (4x16 f32): v0 lanes0-15 = B[0,n], lanes16-31 = B[2,n];
        //                      v1 lanes0-15 = B[1,n], lanes16-31 = B[3,n].
        const float* p = buf + (4 * g + 2 * half) * DIM + 16 * cc + n;
        v2f bmat;
        bmat[0] = p[0];
        bmat[1] = p[DIM];
        acc[cc] = __builtin_amdgcn_wmma_f32_16x16x4_f32(
            false, ones, false, bmat, (short)0, acc[cc], false, false);
      }
    }
  };

  issue(0, g0_0);
  #pragma unroll
  for (int c = 0; c < NCHUNK - 1; ++c) {
    issue(c + 1, ((c + 1) & 1) ? g0_1 : g0_0);
    __builtin_amdgcn_s_wait_tensorcnt(1);       // oldest gather (chunk c) done
    asm volatile("" ::: "memory");  // TDM wrote LDS: block hoisting of ds reads
    consume(c);
    asm volatile("" ::: "memory");
  }
  __builtin_amdgcn_s_wait_tensorcnt(0);         // final gather done
  asm volatile("" ::: "memory");
  consume(NCHUNK - 1);

  // ---- epilogue: every row of D is the column sum; store row 0 (cc 0-3) and row 8 (cc 4-7) ----
  float* orow = out + (size_t)b * DIM;
  #pragma unroll
  for (int cc = 0; cc < 8; ++cc) {
    if ((cc >> 2) == half) orow[cc * 16 + n] = acc[cc][0];
  }
}

extern "C" void kernel_launch(void* const* d_in, const int* in_sizes, int n_in,
                              void* d_out, int out_size, void* d_ws, size_t ws_size,
                              hipStream_t stream) {
  const uint32_t* x32   = (const uint32_t*)d_in[0];   // int64 ids, LE word pairs
  const float*    table = (const float*)d_in[1];
  float*          outp  = (float*)d_out;

  const uint32_t batch = (uint32_t)(in_sizes[0] / HIST);
  const uint32_t nrows = (uint32_t)(in_sizes[1] / DIM);

  const uint32_t nblocks = (batch + WPB - 1) / WPB;
  hipLaunchKernelGGL(hashemb_tdm_wmma_kernel, dim3(nblocks), dim3(32 * WPB), 0,
                     stream, x32, table, outp, nrows, batch);
}